// ChamferLoss_46136538694019
// MI455X (gfx1250) — compile-verified
//
#include <hip/hip_runtime.h>
#include <cfloat>

// CDNA5 / gfx1250 Chamfer loss.
// d2[b,n,m] = |p|^2 + |v|^2 - 2 p.v ; min over m ; mean over n.
// Inner product computed with V_WMMA_F32_16X16X4_F32:
//   A (16x4)  row = [-2px, -2py, -2pz, 1]       (per source point)
//   B (4x16)  col = [ vx ,  vy ,  vz , |v|^2]   (per target vertex)
//   D = A*B = |v|^2 - 2 p.v   (per 16x16 tile) ; |p|^2 added after the min.
// Targets staged in LDS pre-swizzled into the WMMA B-operand layout:
//   s[2m] = {vx,vy}  (K0,K1 rows -> lanes 0-15)
//   s[2m+1] = {vz,v2} (K2,K3 rows -> lanes 16-31)
// so each lane issues one conflict-free ds_load_b64 per 16-target chunk.

typedef __attribute__((ext_vector_type(2))) float v2f;
typedef __attribute__((ext_vector_type(8))) float v8f;

constexpr int B_ = 4;
constexpr int N_ = 16384;
constexpr int M_ = 4096;
constexpr int WAVES = 8;                               // waves per block (wave32)
constexpr int ROWS_PER_BLOCK = WAVES * 16;             // 128 source points / block
constexpr int BLOCKS_PER_BATCH = N_ / ROWS_PER_BLOCK;  // 128
constexpr int TOTAL_BLOCKS = B_ * BLOCKS_PER_BATCH;    // 512
constexpr int TOTAL_WAVES = TOTAL_BLOCKS * WAVES;      // 4096 partials

__global__ void __launch_bounds__(256)
chamfer_min_kernel(const float* __restrict__ src,
                   const float* __restrict__ tgt,
                   float* __restrict__ partials,   // TOTAL_WAVES floats, or nullptr
                   float* __restrict__ out,        // atomic fallback target
                   float invN)
{
    // 4096 targets * 2 float2 = exactly 64KB of LDS; row-scratch reuses it later.
    __shared__ float2 s_tv[2 * M_];

    const int tid  = threadIdx.x;
    const int bidx = blockIdx.x;
    const int b    = bidx / BLOCKS_PER_BATCH;
    const int blk  = bidx % BLOCKS_PER_BATCH;

    // ---- stage this batch's targets, pre-swizzled for the B operand ----
    const float* tb = tgt + (size_t)b * M_ * 3;
    for (int m = tid; m < M_; m += 256) {
        float vx = tb[3 * m + 0];
        float vy = tb[3 * m + 1];
        float vz = tb[3 * m + 2];
        s_tv[2 * m + 0] = make_float2(vx, vy);                      // K0, K1
        s_tv[2 * m + 1] = make_float2(vz, vx * vx + vy * vy + vz * vz); // K2, K3=|v|^2
    }
    __syncthreads();

    const int  wave = tid >> 5;
    const int  lane = tid & 31;
    const int  col  = lane & 15;
    const bool lo   = lane < 16;

    // ---- A operand: 16 source points for this wave ----
    // A 16x4 f32 layout: lanes 0-15 hold K=0/K=1 of row=lane,
    //                    lanes 16-31 hold K=2/K=3 of row=lane-16.
    const int rowBase = blk * ROWS_PER_BLOCK + wave * 16;
    const float* sp = src + ((size_t)b * N_ + (rowBase + col)) * 3;
    const float px = sp[0], py = sp[1], pz = sp[2];
    const float p2 = px * px + py * py + pz * pz;   // row (lane&15)'s |p|^2

    v2f a;
    a.x = lo ? (-2.0f * px) : (-2.0f * pz);
    a.y = lo ? (-2.0f * py) : 1.0f;

    // Per-lane B-operand slot: element (2*col + half) of each 32-float2 chunk row.
    const int bslot = 2 * col + (lo ? 0 : 1);

    float mn[8];
#pragma unroll
    for (int r = 0; r < 8; ++r) mn[r] = FLT_MAX;

    // ---- sweep targets, 16 at a time: 1 ds_load_b64 + 1 WMMA + 8 mins ----
#pragma unroll 4
    for (int c = 0; c < M_ / 16; ++c) {
        float2 t = s_tv[c * 32 + bslot];
        v2f bb;
        bb.x = t.x;
        bb.y = t.y;
        v8f acc = {};                   // C = 0
        acc = __builtin_amdgcn_wmma_f32_16x16x4_f32(
            /*neg_a=*/false, a, /*neg_b=*/false, bb,
            /*c_mod=*/(short)0, acc, /*reuse_a=*/false, /*reuse_b=*/false);
#pragma unroll
        for (int r = 0; r < 8; ++r) mn[r] = fminf(mn[r], acc[r]);
    }

    // ---- min across the 16 lanes of each half (columns) ----
#pragma unroll
    for (int r = 0; r < 8; ++r) {
#pragma unroll
        for (int off = 1; off < 16; off <<= 1)
            mn[r] = fminf(mn[r], __shfl_xor(mn[r], off, 32));
    }

    // ---- per-row results via LDS (reuse target buffer after all reads done) ----
    __syncthreads();
    float* s_row = reinterpret_cast<float*>(s_tv);   // 8 waves * 16 floats
    if (col == 0) {
#pragma unroll
        for (int r = 0; r < 8; ++r)
            s_row[wave * 16 + (lo ? r : r + 8)] = mn[r];  // half0: rows 0-7, half1: 8-15
    }
    __syncthreads();

    // lanes 0..15 own rows 0..15; add their own |p|^2 and clamp at 0.
    float d = 0.0f;
    if (lo) d = fmaxf(s_row[wave * 16 + col] + p2, 0.0f);

    // sum the 16 per-row values (butterfly stays inside lanes 0..15)
#pragma unroll
    for (int off = 1; off < 16; off <<= 1)
        d += __shfl_xor(d, off, 32);

    if (lane == 0) {
        if (partials) partials[bidx * WAVES + wave] = d;
        else          atomicAdd(&out[b], d * invN);
    }
}

__global__ void __launch_bounds__(256)
chamfer_reduce_kernel(const float* __restrict__ partials,
                      float* __restrict__ out, float invN)
{
    __shared__ float sdata[256];
    const int b = blockIdx.x;
    const float* p = partials + (size_t)b * (BLOCKS_PER_BATCH * WAVES);
    float s = 0.0f;
    for (int i = threadIdx.x; i < BLOCKS_PER_BATCH * WAVES; i += 256) s += p[i];
    sdata[threadIdx.x] = s;
    __syncthreads();
    for (int st = 128; st > 0; st >>= 1) {
        if (threadIdx.x < st) sdata[threadIdx.x] += sdata[threadIdx.x + st];
        __syncthreads();
    }
    if (threadIdx.x == 0) out[b] = sdata[0] * invN;
}

__global__ void zero_kernel(float* out, int n)
{
    int i = blockIdx.x * blockDim.x + threadIdx.x;
    if (i < n) out[i] = 0.0f;
}

extern "C" void kernel_launch(void* const* d_in, const int* in_sizes, int n_in,
                              void* d_out, int out_size, void* d_ws, size_t ws_size,
                              hipStream_t stream)
{
    const float* src = (const float*)d_in[0];   // (B, N, 3) fp32
    const float* tgt = (const float*)d_in[1];   // (B, M, 3) fp32
    float* out = (float*)d_out;                 // (B,) fp32
    const float invN = 1.0f / (float)N_;

    const size_t needed = (size_t)TOTAL_WAVES * sizeof(float);
    if (ws_size >= needed) {
        float* partials = (float*)d_ws;
        chamfer_min_kernel<<<TOTAL_BLOCKS, 256, 0, stream>>>(src, tgt, partials, nullptr, invN);
        chamfer_reduce_kernel<<<B_, 256, 0, stream>>>(partials, out, invN);
    } else {
        zero_kernel<<<1, 32, 0, stream>>>(out, out_size);
        chamfer_min_kernel<<<TOTAL_BLOCKS, 256, 0, stream>>>(src, tgt, nullptr, out, invN);
    }
}